// Sober_13288628814379
// MI455X (gfx1250) — compile-verified
//
#include <hip/hip_runtime.h>
#include <cstdint>

// ---------------- problem constants (from reference) ----------------
#define NC    64           // N*C = 2*32
#define DD    64
#define HH    128
#define WW    128
#define PLANE (HH*WW)

// ---------------- tiling ----------------
#define HT    16
#define WT    64
#define LROWS (HT + 2)          // 18 rows (h halo)
#define LCOLS (WT + 2)          // 66 cols (w halo) == LDS row stride (66%64=2 -> bank-conflict-free)
#define TILE  (LROWS * LCOLS)   // 1188 elements per plane tile
#define NLD   ((TILE + 255) / 256) // 5 load slots per thread

typedef float v4f __attribute__((ext_vector_type(4)));

// CDNA5 async DMA: copy 4B from global (SGPR64 base + VGPR byte offset) to LDS
// (VGPR byte address). Tracked with ASYNCcnt.
__device__ __forceinline__ void async_b32(unsigned lds_addr, unsigned voff,
                                          const float* sbase) {
  asm volatile("global_load_async_to_lds_b32 %0, %1, %2"
               :: "v"(lds_addr), "v"(voff), "s"(sbase)
               : "memory");
}
__device__ __forceinline__ void wait_async0() {
  asm volatile("s_wait_asynccnt 0x0" ::: "memory");
}

__global__ void __launch_bounds__(256)
sobel3d_mag(const float* __restrict__ x, float* __restrict__ out) {
  __shared__ float lds[2][TILE];

  const int tid = threadIdx.x;
  const int tw  = tid & 15;   // 0..15 -> quad of 4 w-outputs
  const int th  = tid >> 4;   // 0..15 -> h row
  const int w0  = blockIdx.x * WT;
  const int h0  = blockIdx.y * HT;
  const long long ncbase = (long long)blockIdx.z * (long long)(DD * PLANE);
  const float* __restrict__ xb = x + ncbase;
  float* __restrict__ ob = out + ncbase;

  // ---- plane-invariant per-thread load slots ----
  unsigned ldsoff[NLD], goff[NLD];
  bool vld[NLD];
  const unsigned lds0 = (unsigned)(uintptr_t)(&lds[0][0]); // low 32 bits of a
                                                           // generic LDS ptr ==
                                                           // DS byte address
  #pragma unroll
  for (int it = 0; it < NLD; ++it) {
    int e = tid + it * 256;
    bool act = (e < TILE);
    int ee = act ? e : 0;
    int i = ee / LCOLS;          // local row (h = h0-1+i)
    int j = ee - i * LCOLS;      // local col (w = w0-1+j)
    int h = h0 - 1 + i;
    int w = w0 - 1 + j;
    vld[it]    = act && (h >= 0) && (h < HH) && (w >= 0) && (w < WW);
    ldsoff[it] = (unsigned)ee * 4u;
    goff[it]   = (unsigned)(h * WW + w) * 4u;   // garbage if !vld (unused)
  }

  // zero-pad halo slots once (async loads never touch them afterwards)
  #pragma unroll
  for (int it = 0; it < NLD; ++it) {
    int e = tid + it * 256;
    if (e < TILE && !vld[it]) { lds[0][e] = 0.0f; lds[1][e] = 0.0f; }
  }
  __syncthreads();

  auto issue_plane = [&](int buf, int p) {
    const unsigned pb = (unsigned)(p * PLANE) * 4u;
    const unsigned lb = lds0 + (unsigned)(buf * TILE) * 4u;
    #pragma unroll
    for (int it = 0; it < NLD; ++it)
      if (vld[it]) async_b32(lb + ldsoff[it], pb + goff[it], xb);
  };

  // In-plane separable partials from the tile in LDS:
  //   P1 = (Sh (x) Dw) * plane,  P2 = (Dh (x) Sw) * plane,  P3 = (Sh (x) Sw) * plane
  auto plane_P = [&](int buf, v4f& p1, v4f& p2, v4f& p3) {
    const float* B = &lds[buf][0] + th * LCOLS + tw * 4;
    float rr[3][4], cc[3][4];
    #pragma unroll
    for (int m = 0; m < 3; ++m) {
      const float* row = B + m * LCOLS;
      float a0 = row[0], a1 = row[1], a2 = row[2];
      float a3 = row[3], a4 = row[4], a5 = row[5];
      rr[m][0] = a0 + 2.0f * a1 + a2;  cc[m][0] = a0 - a2;
      rr[m][1] = a1 + 2.0f * a2 + a3;  cc[m][1] = a1 - a3;
      rr[m][2] = a2 + 2.0f * a3 + a4;  cc[m][2] = a2 - a4;
      rr[m][3] = a3 + 2.0f * a4 + a5;  cc[m][3] = a3 - a5;
    }
    #pragma unroll
    for (int k = 0; k < 4; ++k) {
      p1[k] = cc[0][k] + 2.0f * cc[1][k] + cc[2][k];
      p2[k] = rr[0][k] - rr[2][k];
      p3[k] = rr[0][k] + 2.0f * rr[1][k] + rr[2][k];
    }
  };

  // ---- software pipeline over depth ----
  issue_plane(0, 0);
  wait_async0();
  __syncthreads();
  issue_plane(1, 1);

  v4f z = {0.0f, 0.0f, 0.0f, 0.0f};
  v4f p1p = z, p2p = z, p3p = z;       // P(d-1) ; plane -1 is zero padding
  v4f p1c, p2c, p3c;                   // P(d)
  plane_P(0, p1c, p2c, p3c);

  const int orow = (h0 + th) * WW + (w0 + tw * 4);
  for (int d = 0; d < DD; ++d) {
    v4f p1n, p2n, p3n;                 // P(d+1)
    if (d + 1 < DD) {
      wait_async0();                   // my async loads of plane d+1 done
      __syncthreads();                 // all waves' loads done; all reads of
                                       // buf[d&1] (plane d) are behind us
      if (d + 2 < DD) issue_plane(d & 1, d + 2);  // overwrite retired buffer
      plane_P((d + 1) & 1, p1n, p2n, p3n);
    } else {
      p1n = z; p2n = z; p3n = z;       // plane 64 is zero padding
    }

    v4f gx = p1p + 2.0f * p1c + p1n;   // Sz over P1
    v4f gy = p2p + 2.0f * p2c + p2n;   // Sz over P2
    v4f gz = p3p - p3n;                // Dz over P3
    v4f m2 = gx * gx + gy * gy + gz * gz + 1e-6f;
    v4f o;
    #pragma unroll
    for (int k = 0; k < 4; ++k) o[k] = __builtin_sqrtf(m2[k]);
    __builtin_nontemporal_store(o, (v4f*)(ob + (size_t)d * PLANE + orow));

    p1p = p1c; p2p = p2c; p3p = p3c;
    p1c = p1n; p2c = p2n; p3c = p3n;
  }
}

extern "C" void kernel_launch(void* const* d_in, const int* in_sizes, int n_in,
                              void* d_out, int out_size, void* d_ws, size_t ws_size,
                              hipStream_t stream) {
  (void)in_sizes; (void)n_in; (void)d_ws; (void)ws_size; (void)out_size;
  const float* x = (const float*)d_in[0];   // kernels d_in[1..3] are the fixed
                                            // Sobel taps; hardcoded in-kernel.
  float* out = (float*)d_out;
  dim3 grid(WW / WT, HH / HT, NC);          // (2, 8, 64) = 1024 blocks
  sobel3d_mag<<<grid, 256, 0, stream>>>(x, out);
}